// SpikeLoss_3831110828785
// MI455X (gfx1250) — compile-verified
//
#include <hip/hip_runtime.h>

// SpikeLoss: x = outputs - target; causal EMA along T (decay 0.75); loss = 0.5*sum((syn/tau)^2)
// EMA-scan-as-triangular-matmul on v_wmma_f32_16x16x4_f32, carry via ds_bpermute.
// Bandwidth-bound: 256 MiB in / 23.3 TB/s ~= 11.5 us floor; WMMA math is negligible.
// Data path: 4x global_load_b128 issued back-to-back -> v_pk_add_f32(neg) -> ds_store_b128
// into wave-private double-buffered LDS tile -> 8x ds_load_b32 B-fragments (issued ahead
// of the WMMA chain) -> 4 chained WMMAs. Same-wave DS ops are in-order: no barriers.

typedef __attribute__((ext_vector_type(2))) float v2f;
typedef __attribute__((ext_vector_type(8))) float v8f;

#define TAU_S 4.0f
// DECAY = 0.75; DECAY^8 = 6561/65536 (exact in fp32)
#define DECAY8 0.1001129150390625f

#define TSTR    20            // LDS tile row stride in dwords (16B-aligned rows, bank-spread)
#define TILE_DW (16 * TSTR)   // dwords per tile buffer

// DECAY^e for e in [0,31]; exact for e<=15 since 3^e < 2^24.
__device__ __forceinline__ float dpow(int e) {
  float p = 1.0f;
  p *= (e & 1) ? 0.75f : 1.0f;
  p *= (e & 2) ? 0.5625f : 1.0f;
  p *= (e & 4) ? 0.31640625f : 1.0f;
  p *= (e & 8) ? DECAY8 : 1.0f;
  p *= (e & 16) ? 1.00225959904491901397e-2f : 1.0f;
  return p;
}

__global__ __launch_bounds__(256) void spike_scan_wmma(
    const float* __restrict__ outp,
    const float* __restrict__ targ,
    float* __restrict__ partials,
    int cols)  // cols = B*N, T fixed at 128
{
  __shared__ float lds[8 * 2 * TILE_DW];  // 8 waves x 2 buffers x 1280B = 20KB

  const int lane = threadIdx.x & 31;
  const int wave = threadIdx.x >> 5;
  const int half = lane >> 4;   // 0: lanes 0-15, 1: lanes 16-31
  const int n    = lane & 15;   // column-in-tile (B/C/D) and row M (A)
  const int col0 = blockIdx.x * 128 + wave * 16;

  float* const wtile = &lds[wave * 2 * TILE_DW];

  // Loader mapping: lane -> (row, 4-col group) of the 16x16 chunk tile.
  const int lrow = lane >> 2;        // 0..7  (+8 for second b128)
  const int lcol = (lane & 3) * 4;   // 0,4,8,12

  // Strength-reduced streaming pointers: one uniform bump of 16*cols per chunk.
  const long rbase = (long)lrow * cols + col0 + lcol;
  const long rstep = 8L * cols;      // p=0 -> p=1 row offset
  const long cstep = 16L * cols;     // chunk stride
  const float* po0 = outp + rbase;
  const float* po1 = po0 + rstep;
  const float* pt0 = targ + rbase;
  const float* pt1 = pt0 + rstep;

  // A fragments: lower-triangular Lc[m][k] = DECAY^(m-k), k<=m (16x16, K split in 4 groups).
  // f32 16x16x4 A layout: lane holds M=lane&15; VGPR0 -> K = 4j + 2*half, VGPR1 -> K+1.
  v2f a[4];
#pragma unroll
  for (int j = 0; j < 4; ++j) {
    const int k0 = 4 * j + 2 * half;
    a[j].x = (n >= k0)     ? dpow(n - k0)     : 0.0f;
    a[j].y = (n >= k0 + 1) ? dpow(n - k0 - 1) : 0.0f;
  }

  const float hfac = half ? DECAY8 : 1.0f;  // DECAY^(8*half)

  float carry = 0.0f;   // syn state of column col0+n entering the chunk (dup in both halves)
  float acc   = 0.0f;   // per-lane sum of syn^2 over owned D elements

#pragma unroll
  for (int chunk = 0; chunk < 8; ++chunk) {
    float* const buf = wtile + (chunk & 1) * TILE_DW;

    // Issue all 4 global b128 loads before consuming any of them.
    const float4 o0 = *(const float4*)po0;
    const float4 o1 = *(const float4*)po1;
    const float4 g0 = *(const float4*)pt0;
    const float4 g1 = *(const float4*)pt1;
    po0 += cstep; po1 += cstep; pt0 += cstep; pt1 += cstep;

    float4 x0, x1;
    x0.x = o0.x - g0.x;  x0.y = o0.y - g0.y;  x0.z = o0.z - g0.z;  x0.w = o0.w - g0.w;
    x1.x = o1.x - g1.x;  x1.y = o1.y - g1.y;  x1.z = o1.z - g1.z;  x1.w = o1.w - g1.w;
    *(float4*)(buf + lrow * TSTR + lcol)       = x0;
    *(float4*)(buf + (lrow + 8) * TSTR + lcol) = x1;

    // Fetch all 4 B fragments ahead of the WMMA chain (staged dscnt waits).
    v2f b[4];
#pragma unroll
    for (int j = 0; j < 4; ++j) {
      const int k0 = 4 * j + 2 * half;
      b[j].x = buf[k0 * TSTR + n];
      b[j].y = buf[(k0 + 1) * TSTR + n];
    }

    // C init: S[m][n] = DECAY^(m+1) * carry[n],  m = r + 8*half
    const float hc = hfac * carry;
    v8f d;
#pragma unroll
    for (int r = 0; r < 8; ++r)
      d[r] = dpow(r + 1) * hc;

#pragma unroll
    for (int j = 0; j < 4; ++j)
      d = __builtin_amdgcn_wmma_f32_16x16x4_f32(false, a[j], false, b[j],
                                                (short)0, d, false, false);

#pragma unroll
    for (int r = 0; r < 8; ++r) acc += d[r] * d[r];

    // New carry = S[15][n] = d[7] held by lane n+16; broadcast to both half-waves.
    carry = __int_as_float(
        __builtin_amdgcn_ds_bpermute((lane | 16) << 2, __float_as_int(d[7])));
  }

  // Block reduction -> one partial per block (deterministic tree).
  __shared__ float red[256];
  red[threadIdx.x] = acc;
  __syncthreads();
#pragma unroll
  for (int s = 128; s > 0; s >>= 1) {
    if ((int)threadIdx.x < s) red[threadIdx.x] += red[threadIdx.x + s];
    __syncthreads();
  }
  if (threadIdx.x == 0) partials[blockIdx.x] = red[0];
}

__global__ __launch_bounds__(256) void spike_finalize(
    const float* __restrict__ partials, int nPartials, float* __restrict__ out)
{
  __shared__ float red[256];
  float s = 0.0f;
  for (int i = threadIdx.x; i < nPartials; i += 256) s += partials[i];  // fixed order
  red[threadIdx.x] = s;
  __syncthreads();
#pragma unroll
  for (int st = 128; st > 0; st >>= 1) {
    if ((int)threadIdx.x < st) red[threadIdx.x] += red[threadIdx.x + st];
    __syncthreads();
  }
  if (threadIdx.x == 0) out[0] = (0.5f / (TAU_S * TAU_S)) * red[0];
}

extern "C" void kernel_launch(void* const* d_in, const int* in_sizes, int n_in,
                              void* d_out, int out_size, void* d_ws, size_t ws_size,
                              hipStream_t stream) {
  const float* outputs = (const float*)d_in[0];
  const float* target  = (const float*)d_in[1];
  float* out = (float*)d_out;

  const int T = 128;
  const long total = (long)in_sizes[0];
  const int cols = (int)(total / T);     // 256*1024 = 262144
  const int blocks = cols / 128;         // 2048 (exact)

  float* partials = (float*)d_ws;        // blocks * 4 bytes scratch

  spike_scan_wmma<<<blocks, 256, 0, stream>>>(outputs, target, partials, cols);
  spike_finalize<<<1, 256, 0, stream>>>(partials, blocks, out);
}